// GCN_Net1_1529008358070
// MI455X (gfx1250) — compile-verified
//
#include <hip/hip_runtime.h>

#define N_NODES 50000
#define N_EDGES 800000
#define N_GRAPH 64
#define BN_EPS  1e-5f

typedef __attribute__((ext_vector_type(2))) float v2f;
typedef __attribute__((ext_vector_type(8))) float v8f;

// ---------------------------------------------------------------- utilities
__global__ void fill_zero_kernel(float* __restrict__ p, int n) {
  int i = blockIdx.x * blockDim.x + threadIdx.x;
  if (i < n) p[i] = 0.0f;
}

// deg = #incoming edges per node (self-loop added later as +1)
__global__ void deg_kernel(const int* __restrict__ col, float* __restrict__ deg, int E) {
  int e = blockIdx.x * blockDim.x + threadIdx.x;
  if (e < E) atomicAdd(&deg[col[e]], 1.0f);
}

__global__ void dinv_kernel(const float* __restrict__ deg, float* __restrict__ dinv, int n) {
  int i = blockIdx.x * blockDim.x + threadIdx.x;
  if (i < n) dinv[i] = rsqrtf(deg[i] + 1.0f);   // +1 = self loop; always > 0
}

// layer 1 has din == 1: h[n][j] = x[n] * W1[0][j]
__global__ void layer1_kernel(const float* __restrict__ x, const float* __restrict__ W1,
                              float* __restrict__ h, int n) {
  int idx = blockIdx.x * blockDim.x + threadIdx.x;
  if (idx < n * 64) h[idx] = x[idx >> 6] * W1[idx & 63];
}

// ------------------------------------------------- H[n,64] = X[n,64] @ W[64,64]
// f32 WMMA path: V_WMMA_F32_16X16X4_F32.  One wave -> 16x64 output strip.
// A 16x4 layout: lane l (l<16): M=l, K in {k,k+1}; lane l>=16: M=l-16, K in {k+2,k+3}
// D 16x16 layout: VGPR r -> M = r + 8*(lane/16), N = lane%16
//
// W is staged into LDS in *fragment order*: for fragment (kk, nt), lane l holds
// the float2 {W[kk*4+2*(l/16)+0][nt*16+l%16], W[kk*4+2*(l/16)+1][...]} at a
// contiguous 8-byte LDS address -> a single aligned ds_load_b64 per B operand,
// no VGPR repacking in the inner loop.
__global__ void gemm_n64_wmma(const float* __restrict__ X, const float* __restrict__ W,
                              float* __restrict__ H, int nrows) {
  __shared__ float sW[64 * 64];                 // swizzled copy of W
  for (int i = threadIdx.x; i < 64 * 64; i += blockDim.x) {
    const int kr = i >> 6, nc = i & 63;         // source: W[kr][nc] (coalesced read)
    const int kk = kr >> 2, hf = (kr >> 1) & 1, j = kr & 1;
    const int nt = nc >> 4, mm = nc & 15;
    const int ln = hf * 16 + mm;
    sW[((kk * 4 + nt) * 32 + ln) * 2 + j] = W[i];
  }
  __syncthreads();

  const int wave = threadIdx.x >> 5;
  const int lane = threadIdx.x & 31;
  const int tile = blockIdx.x * (blockDim.x >> 5) + wave;
  const int ntiles = (nrows + 15) >> 4;
  if (tile >= ntiles) return;                   // wave-uniform exit, EXEC stays full

  const int m    = lane & 15;
  const int half = lane >> 4;                   // 0: K pair base +0, 1: +2
  const int row0 = tile << 4;
  int row = row0 + m;
  if (row >= nrows) row = nrows - 1;            // N is a multiple of 16: no-op

  v8f acc[4] = {v8f{}, v8f{}, v8f{}, v8f{}};

#pragma unroll
  for (int kk = 0; kk < 16; ++kk) {
    v2f a = *(const v2f*)(X + row * 64 + kk * 4 + 2 * half);   // global_load_b64
#pragma unroll
    for (int nt = 0; nt < 4; ++nt) {
      v2f b = *(const v2f*)(sW + ((kk * 4 + nt) * 32 + lane) * 2);  // ds_load_b64
      acc[nt] = __builtin_amdgcn_wmma_f32_16x16x4_f32(
          false, a, false, b, (short)0, acc[nt], false, false);
    }
  }

#pragma unroll
  for (int nt = 0; nt < 4; ++nt) {
#pragma unroll
    for (int r = 0; r < 8; ++r) {
      int orow = row0 + r + 8 * half;
      if (orow < nrows) H[orow * 64 + nt * 16 + m] = acc[nt][r];
    }
  }
}

// ---------------------------- fused gather * norm -> scatter-add (incl. self loops)
// one wave per edge; 32 lanes x 2 channels; norm recomputed from dinv
__global__ void edge_scatter_kernel(const float* __restrict__ H, const int* __restrict__ row,
                                    const int* __restrict__ col, const float* __restrict__ dinv,
                                    float* __restrict__ agg, int E, int n) {
  const int e    = blockIdx.x * (blockDim.x >> 5) + (threadIdx.x >> 5);
  const int lane = threadIdx.x & 31;
  if (e >= E + n) return;
  int r, c; float nrm;
  if (e < E) { r = row[e]; c = col[e]; nrm = dinv[r] * dinv[c]; }
  else       { r = e - E;  c = r;      float d = dinv[r]; nrm = d * d; }
  atomicAdd(&agg[c * 64 + lane],      H[r * 64 + lane]      * nrm);
  atomicAdd(&agg[c * 64 + lane + 32], H[r * 64 + lane + 32] * nrm);
}

// ------------------------------------------------------------ BatchNorm stats
__global__ void bn_stats_kernel(const float* __restrict__ agg, float* __restrict__ sums,
                                float* __restrict__ sumsq, int n) {
  __shared__ float ssum[256], ssq[256];
  const int c  = threadIdx.x & 63;
  const int rg = threadIdx.x >> 6;              // 4 row-groups per block
  float s = 0.f, q = 0.f;
  for (int i = blockIdx.x * 4 + rg; i < n; i += gridDim.x * 4) {
    float v = agg[i * 64 + c];
    s += v; q += v * v;
  }
  ssum[threadIdx.x] = s; ssq[threadIdx.x] = q;
  __syncthreads();
  if (rg == 0) {
    s = ssum[c] + ssum[c + 64] + ssum[c + 128] + ssum[c + 192];
    q = ssq[c]  + ssq[c + 64]  + ssq[c + 128]  + ssq[c + 192];
    atomicAdd(&sums[c], s);
    atomicAdd(&sumsq[c], q);
  }
}

// BN + ReLU (pre-BN bias b cancels exactly inside BN, so it is skipped)
__global__ void bn_apply_relu_kernel(const float* __restrict__ agg, const float* __restrict__ sums,
                                     const float* __restrict__ sumsq, const float* __restrict__ g,
                                     const float* __restrict__ bt, float* __restrict__ out, int n) {
  int idx = blockIdx.x * blockDim.x + threadIdx.x;
  if (idx >= n * 64) return;
  int c = idx & 63;
  float inv  = 1.0f / (float)n;
  float mean = sums[c] * inv;
  float var  = sumsq[c] * inv - mean * mean;
  float v = (agg[idx] - mean) * rsqrtf(var + BN_EPS);
  v = g[c] * v + bt[c];
  out[idx] = v > 0.f ? v : 0.f;
}

// -------------------------------------------------------------- mean pooling
__global__ void pool_kernel(const float* __restrict__ act, const int* __restrict__ batch,
                            float* __restrict__ pooled, float* __restrict__ cnt, int n) {
  int idx = blockIdx.x * blockDim.x + threadIdx.x;
  if (idx >= n * 64) return;
  int node = idx >> 6, c = idx & 63;
  int gph = batch[node];
  atomicAdd(&pooled[gph * 64 + c], act[idx]);
  if (c == 0) atomicAdd(&cnt[gph], 1.0f);
}

// ----------------------------------------------------------------- MLP head
__global__ void mlp_kernel(const float* __restrict__ pooled, const float* __restrict__ cnt,
                           const float* __restrict__ fw1, const float* __restrict__ fb1,
                           const float* __restrict__ fw2, const float* __restrict__ fb2,
                           float* __restrict__ out) {
  __shared__ float p[64], hidden[128];
  int g = blockIdx.x, t = threadIdx.x;
  if (t < 64) {
    float c = cnt[g]; if (c < 1.f) c = 1.f;
    p[t] = pooled[g * 64 + t] / c;
  }
  __syncthreads();
  float s = fb1[t];
  for (int k = 0; k < 64; ++k) s += p[k] * fw1[t * 64 + k];
  hidden[t] = s > 0.f ? s : 0.f;
  __syncthreads();
  if (t < 10) {
    float o = fb2[t];
    for (int k = 0; k < 128; ++k) o += hidden[k] * fw2[t * 128 + k];
    out[g * 10 + t] = o;
  }
}

// ---------------------------------------------------------------------------
extern "C" void kernel_launch(void* const* d_in, const int* in_sizes, int n_in,
                              void* d_out, int out_size, void* d_ws, size_t ws_size,
                              hipStream_t stream) {
  (void)in_sizes; (void)n_in; (void)out_size; (void)ws_size;

  const float* x     = (const float*)d_in[0];
  const int*   eidx  = (const int*)d_in[1];
  const int*   batch = (const int*)d_in[2];
  const float* W[4]  = {(const float*)d_in[3],  (const float*)d_in[7],
                        (const float*)d_in[11], (const float*)d_in[15]};
  const float* gg[4] = {(const float*)d_in[5],  (const float*)d_in[9],
                        (const float*)d_in[13], (const float*)d_in[17]};
  const float* bt[4] = {(const float*)d_in[6],  (const float*)d_in[10],
                        (const float*)d_in[14], (const float*)d_in[18]};
  const float* fw1 = (const float*)d_in[19];
  const float* fb1 = (const float*)d_in[20];
  const float* fw2 = (const float*)d_in[21];
  const float* fb2 = (const float*)d_in[22];
  const int* row = eidx;
  const int* col = eidx + N_EDGES;

  // workspace carve-up (256B aligned)
  char* ws = (char*)d_ws;
  size_t off = 0;
  auto carve = [&](size_t bytes) {
    void* p = ws + off;
    off += (bytes + 255) & ~(size_t)255;
    return p;
  };
  float* bufA   = (float*)carve(sizeof(float) * N_NODES * 64);  // activations
  float* bufB   = (float*)carve(sizeof(float) * N_NODES * 64);  // aggregation
  float* bufC   = (float*)carve(sizeof(float) * N_NODES * 64);  // gemm output h
  float* deg    = (float*)carve(sizeof(float) * N_NODES);
  float* dinv   = (float*)carve(sizeof(float) * N_NODES);
  float* stats  = (float*)carve(sizeof(float) * 128);           // sums | sumsq
  float* pooled = (float*)carve(sizeof(float) * (N_GRAPH * 64 + N_GRAPH)); // sums | cnt
  float* sums   = stats;
  float* sumsq  = stats + 64;
  float* cnt    = pooled + N_GRAPH * 64;

  const int TPB = 256;
#define GRID(n) ((((n) + TPB - 1) / TPB))

  // degree / normalization
  fill_zero_kernel<<<GRID(N_NODES), TPB, 0, stream>>>(deg, N_NODES);
  deg_kernel<<<GRID(N_EDGES), TPB, 0, stream>>>(col, deg, N_EDGES);
  dinv_kernel<<<GRID(N_NODES), TPB, 0, stream>>>(deg, dinv, N_NODES);

  const int gemm_blocks    = (((N_NODES + 15) / 16) + 7) / 8;        // 8 waves/block
  const int scatter_blocks = ((N_EDGES + N_NODES) + 7) / 8;          // 8 edges/block

  for (int L = 0; L < 4; ++L) {
    if (L == 0) {
      layer1_kernel<<<GRID(N_NODES * 64), TPB, 0, stream>>>(x, W[0], bufC, N_NODES);
    } else {
      gemm_n64_wmma<<<gemm_blocks, TPB, 0, stream>>>(bufA, W[L], bufC, N_NODES);
    }
    fill_zero_kernel<<<GRID(N_NODES * 64), TPB, 0, stream>>>(bufB, N_NODES * 64);
    edge_scatter_kernel<<<scatter_blocks, TPB, 0, stream>>>(bufC, row, col, dinv, bufB,
                                                            N_EDGES, N_NODES);
    fill_zero_kernel<<<1, 128, 0, stream>>>(stats, 128);
    bn_stats_kernel<<<256, TPB, 0, stream>>>(bufB, sums, sumsq, N_NODES);
    bn_apply_relu_kernel<<<GRID(N_NODES * 64), TPB, 0, stream>>>(bufB, sums, sumsq,
                                                                 gg[L], bt[L], bufA, N_NODES);
  }

  // global mean pool + MLP head
  fill_zero_kernel<<<GRID(N_GRAPH * 64 + N_GRAPH), TPB, 0, stream>>>(pooled,
                                                                     N_GRAPH * 64 + N_GRAPH);
  pool_kernel<<<GRID(N_NODES * 64), TPB, 0, stream>>>(bufA, batch, pooled, cnt, N_NODES);
  mlp_kernel<<<N_GRAPH, 128, 0, stream>>>(pooled, cnt, fw1, fb1, fw2, fb2, (float*)d_out);
#undef GRID
}